// FixedEig_80393197846808
// MI455X (gfx1250) — compile-verified
//
#include <hip/hip_runtime.h>
#include <math.h>

// One workgroup (256 threads = 8 wave32) per batch element.
// LDS-resident pipeline: build [a^T|G] -> MGS QR -> q (96x88, zero-padded to
// 96x96) -> T = H*q and h_ = q^T*T via V_WMMA_F32_16X16X4_F32 -> cyclic
// parallel Jacobi eigensolver on 88x88 -> sort -> T2 = q*U via WMMA -> output.

#define NDIM   96
#define NCON   8
#define NMOD   88
#define PITCH  97                  // 97 coprime with 64 banks -> conflict-free columns
#define REG    (NDIM * PITCH)      // 9312 floats per LDS region
#define BLOCK  256
#define NSWEEP 10

typedef __attribute__((ext_vector_type(2))) float v2f;
typedef __attribute__((ext_vector_type(8))) float v8f;

__device__ __forceinline__ float hash_pm1(unsigned x) {
  x ^= x >> 16; x *= 0x7feb352du; x ^= x >> 15; x *= 0x846ca68bu; x ^= x >> 16;
  return (float)(x & 0xFFFFFFu) * (1.0f / 8388608.0f) - 1.0f;  // [-1,1)
}

// C(96x96) = op(A) * B, all pitch-97 LDS matrices, K = 96.
// Tiles 16x16, K-step 4, one V_WMMA_F32_16X16X4_F32 per step.
// A 16x4 f32 layout: lane half h, sublane l: A(M=l, K=2h+v) in component v.
// B 4x16 layout mirrors: component v holds B(K=2h+v, N=l).
// C layout: component r holds C(M=r+8h, N=l).
__device__ __forceinline__ void gemm96_wmma(const float* __restrict__ A,
                                            const float* __restrict__ B,
                                            float* __restrict__ C,
                                            bool transA, int tid) {
  const int wave = tid >> 5;
  const int lane = tid & 31;
  const int half = lane >> 4;
  const int l    = lane & 15;
  for (int tile = wave; tile < 36; tile += 8) {          // wave-uniform: EXEC all-1s
    const int M0 = (tile / 6) * 16;
    const int N0 = (tile % 6) * 16;
    v8f acc = {0.f, 0.f, 0.f, 0.f, 0.f, 0.f, 0.f, 0.f};
    for (int k0 = 0; k0 < 96; k0 += 4) {
      const int ka = k0 + 2 * half;
      v2f av, bv;
      if (transA) {
        av.x = A[(ka + 0) * PITCH + (M0 + l)];
        av.y = A[(ka + 1) * PITCH + (M0 + l)];
      } else {
        av.x = A[(M0 + l) * PITCH + (ka + 0)];
        av.y = A[(M0 + l) * PITCH + (ka + 1)];
      }
      bv.x = B[(ka + 0) * PITCH + (N0 + l)];
      bv.y = B[(ka + 1) * PITCH + (N0 + l)];
      acc = __builtin_amdgcn_wmma_f32_16x16x4_f32(false, av, false, bv,
                                                  (short)0, acc, false, false);
    }
#pragma unroll
    for (int r = 0; r < 8; ++r)
      C[(M0 + r + 8 * half) * PITCH + (N0 + l)] = acc[r];
  }
}

__global__ __launch_bounds__(BLOCK)
void fixed_eig_kernel(const float* __restrict__ Hg, const float* __restrict__ Ag,
                      float* __restrict__ Eout, float* __restrict__ Cout) {
  extern __shared__ float smem[];
  float* R0 = smem;            // M -> Q -> q (padded)
  float* R1 = smem + REG;      // H -> h_ -> T2
  float* R2 = smem + 2 * REG;  // temp -> T -> U

  __shared__ float red[128];
  __shared__ float jc[44], js[44];
  __shared__ int   jp[44], jq[44];
  __shared__ float sdiag[NMOD];
  __shared__ int   sperm[NMOD];

  const int b = blockIdx.x;
  const int t = threadIdx.x;
  const float* hB = Hg + (size_t)b * NDIM * NDIM;
  const float* aB = Ag + (size_t)b * NCON * NDIM;

  // ---- Stage 1: build M = [a^T | G] in R0, load H into R1 ----
  for (int idx = t; idx < NDIM * NDIM; idx += BLOCK) {
    int i = idx / NDIM, j = idx % NDIM;
    float v = (j < NCON) ? aB[j * NDIM + i]
                         : hash_pm1((unsigned)(i * NDIM + j) * 2654435761u + 12345u);
    R0[i * PITCH + j] = v;
    R1[i * PITCH + j] = hB[idx];
  }
  __syncthreads();

  // ---- Stage 2: Modified Gram-Schmidt QR on R0 ----
  for (int j = 0; j < NDIM; ++j) {
    if (t < NDIM) red[t] = R0[t * PITCH + j] * R0[t * PITCH + j];
    __syncthreads();
    if (t == 0) {
      float s = 0.f;
      for (int i = 0; i < NDIM; ++i) s += red[i];
      red[0] = rsqrtf(fmaxf(s, 1e-30f));
    }
    __syncthreads();
    float inv = red[0];
    if (t < NDIM) R0[t * PITCH + j] *= inv;
    __syncthreads();
    for (int col = j + 1 + t; col < NDIM; col += BLOCK) {
      float dot = 0.f;
      for (int i = 0; i < NDIM; ++i) dot += R0[i * PITCH + j] * R0[i * PITCH + col];
      for (int i = 0; i < NDIM; ++i) R0[i * PITCH + col] -= dot * R0[i * PITCH + j];
    }
    __syncthreads();
  }

  // ---- Stage 3: q = Q[:, NCON:], zero-padded to 96 cols (via R2 temp) ----
  for (int idx = t; idx < NDIM * NDIM; idx += BLOCK) {
    int i = idx / NDIM, m = idx % NDIM;
    R2[i * PITCH + m] = (m < NMOD) ? R0[i * PITCH + (m + NCON)] : 0.f;
  }
  __syncthreads();
  for (int idx = t; idx < NDIM * NDIM; idx += BLOCK) {
    int i = idx / NDIM, m = idx % NDIM;
    R0[i * PITCH + m] = R2[i * PITCH + m];
  }
  __syncthreads();

  // ---- Stage 4: T = H * q (WMMA), then h_ = q^T * T (WMMA) ----
  gemm96_wmma(R1, R0, R2, false, t);   // R2 = H*q
  __syncthreads();
  gemm96_wmma(R0, R2, R1, true, t);    // R1 = q^T * (H*q); pad rows/cols are 0
  __syncthreads();

  // ---- Stage 5: cyclic parallel Jacobi on R1 (88x88), U in R2 ----
  for (int idx = t; idx < NDIM * NDIM; idx += BLOCK) {
    int i = idx / NDIM, m = idx % NDIM;
    R2[i * PITCH + m] = (i == m && i < NMOD) ? 1.f : 0.f;
  }
  __syncthreads();

  for (int sweep = 0; sweep < NSWEEP; ++sweep) {
    for (int r = 0; r < NMOD - 1; ++r) {
      if (t < 44) {  // round-robin tournament pairing: 44 disjoint pairs
        int p, q;
        if (t == 0) { p = NMOD - 1; q = r % (NMOD - 1); }
        else {
          p = (r + t) % (NMOD - 1);
          q = (r + (NMOD - 1) - t) % (NMOD - 1);
        }
        if (p > q) { int tmp = p; p = q; q = tmp; }
        jp[t] = p; jq[t] = q;
        float app = R1[p * PITCH + p];
        float aqq = R1[q * PITCH + q];
        float apq = R1[p * PITCH + q];
        float c = 1.f, s = 0.f;
        if (fabsf(apq) > 1e-12f) {
          float tau = (aqq - app) / (2.f * apq);
          float tt  = (tau >= 0.f ? 1.f : -1.f) / (fabsf(tau) + sqrtf(1.f + tau * tau));
          c = rsqrtf(1.f + tt * tt);
          s = tt * c;
        }
        jc[t] = c; js[t] = s;
      }
      __syncthreads();
      // row rotations: A <- J^T A  (disjoint pairs -> race-free)
      for (int w = t; w < 44 * NMOD; w += BLOCK) {
        int pr = w / NMOD, j = w % NMOD;
        int p = jp[pr], q = jq[pr];
        float c = jc[pr], s = js[pr];
        float ap = R1[p * PITCH + j], aq = R1[q * PITCH + j];
        R1[p * PITCH + j] = c * ap - s * aq;
        R1[q * PITCH + j] = s * ap + c * aq;
      }
      __syncthreads();
      // column rotations: A <- A J, and U <- U J
      for (int w = t; w < 44 * NMOD; w += BLOCK) {
        int pr = w / NMOD, i = w % NMOD;
        int p = jp[pr], q = jq[pr];
        float c = jc[pr], s = js[pr];
        float ap = R1[i * PITCH + p], aq = R1[i * PITCH + q];
        R1[i * PITCH + p] = c * ap - s * aq;
        R1[i * PITCH + q] = s * ap + c * aq;
        float up = R2[i * PITCH + p], uq = R2[i * PITCH + q];
        R2[i * PITCH + p] = c * up - s * uq;
        R2[i * PITCH + q] = s * up + c * uq;
      }
      __syncthreads();
    }
  }

  // ---- Stage 6: extract diag, compute ascending-order permutation ----
  if (t < NMOD) sdiag[t] = R1[t * PITCH + t];
  __syncthreads();
  if (t < NMOD) {
    float d = sdiag[t];
    int rank = 0;
    for (int k = 0; k < NMOD; ++k) {
      float dk = sdiag[k];
      if (dk < d || (dk == d && k < t)) rank++;
    }
    sperm[rank] = t;
  }
  __syncthreads();

  // ---- Stage 7: T2 = q * U (WMMA); c[k][n] = T2[n][perm[k]] ----
  gemm96_wmma(R0, R2, R1, false, t);   // R1 = q*U (96x88 valid, pad cols garbage-free)
  __syncthreads();

  if (t < NMOD) Eout[(size_t)b * NMOD + t] = sdiag[sperm[t]];
  for (int w = t; w < NMOD * NDIM; w += BLOCK) {
    int k = w / NDIM, n = w % NDIM;
    Cout[(size_t)b * (NMOD * NDIM) + w] = R1[n * PITCH + sperm[k]];
  }
}

extern "C" void kernel_launch(void* const* d_in, const int* in_sizes, int n_in,
                              void* d_out, int out_size, void* d_ws, size_t ws_size,
                              hipStream_t stream) {
  const float* h = (const float*)d_in[0];   // [b, 96, 96] fp32
  const float* a = (const float*)d_in[1];   // [b, 8, 96]  fp32
  const int n_bch = in_sizes[0] / (NDIM * NDIM);
  float* e = (float*)d_out;                       // [b, 88]
  float* c = e + (size_t)n_bch * NMOD;            // [b, 88, 32, 3] flat
  const size_t shmem = (size_t)3 * REG * sizeof(float);  // ~110 KB of 320 KB/WGP
  (void)hipFuncSetAttribute((const void*)fixed_eig_kernel,
                            hipFuncAttributeMaxDynamicSharedMemorySize, (int)shmem);
  hipLaunchKernelGGL(fixed_eig_kernel, dim3(n_bch), dim3(BLOCK), shmem, stream,
                     h, a, e, c);
}